// FSIBlock_74861279969535
// MI455X (gfx1250) — compile-verified
//
#include <hip/hip_runtime.h>
#include <hip/hip_bf16.h>

typedef float v2f __attribute__((ext_vector_type(2)));
typedef float v4f __attribute__((ext_vector_type(4)));
typedef float v8f __attribute__((ext_vector_type(8)));

#define HW_TOTAL 16384   // 128*128
#define NBC      4096    // 16*256
#define CDIM     256
#define HDIM     128
#define BDIM     16

__device__ __forceinline__ float sigmoidf_(float v) {
    return 1.0f / (1.0f + __expf(-v));
}

// ---------------------------------------------------------------------------
// Kernel 1: pooled[b,c] = mean(x[b,c,:,:]).  One block per (b,c) pair.
// Default (RT) temporal policy on the x reads: we deliberately want x lines
// resident in the 192 MB L2 so the reversed second pass can hit them.
// ---------------------------------------------------------------------------
__global__ __launch_bounds__(256) void fsi_pool_kernel(const float* __restrict__ x,
                                                       float* __restrict__ pooled) {
    const int bc = blockIdx.x;
    const int t  = threadIdx.x;
    const v4f* __restrict__ x4 = (const v4f*)(x + (size_t)bc * HW_TOTAL);

    float sum = 0.0f;
#pragma unroll
    for (int i = 0; i < 16; ++i) {
        v4f v = x4[i * 256 + t];             // coalesced: 4 KiB per iteration
        sum += v.x + v.y + v.z + v.w;
    }

    __shared__ float red[256];
    red[t] = sum;
    __syncthreads();
#pragma unroll
    for (int s = 128; s > 0; s >>= 1) {
        if (t < s) red[t] += red[t + s];
        __syncthreads();
    }
    if (t == 0) pooled[bc] = red[0] * (1.0f / (float)HW_TOTAL);
}

// ---------------------------------------------------------------------------
// Kernel 2: tiny MLP + gate via V_WMMA_F32_16X16X4_F32 (M=16 == batch).
//   gate   = sigmoid(pooled @ Wc.T + bc)          (16x256), K=256
//   h      = relu   (pooled @ W1.T + b1)          (16x128), K=256
//   weight = sigmoid(h      @ W2.T + b2)          (16x256), K=128
//   s_out  = weight + gate
// One workgroup, 16 waves; wave w owns N-columns [16w, 16w+16).
// WMMA fragment mapping (wave32, f32):
//   A 16x4 : m = lane%16, VGPR{0,1} = K{kb+2*(lane/16), +1}
//   B 4x16 : n = lane%16, VGPR{0,1} = K{kb+2*(lane/16), +1}
//   C/D    : VGPR v -> m = v + 8*(lane/16), n = lane%16
// ---------------------------------------------------------------------------
__global__ __launch_bounds__(512) void fsi_mlp_kernel(const float* __restrict__ pooled_g,
                                                      const float* __restrict__ W1,
                                                      const float* __restrict__ b1,
                                                      const float* __restrict__ W2,
                                                      const float* __restrict__ b2,
                                                      const float* __restrict__ Wc,
                                                      const float* __restrict__ bcv,
                                                      float* __restrict__ s_out) {
    __shared__ float pooled[BDIM * CDIM];   // 16 KiB
    __shared__ float hbuf[BDIM * HDIM];     //  8 KiB

    const int tid = threadIdx.x;
    for (int i = tid; i < BDIM * CDIM; i += 512) pooled[i] = pooled_g[i];
    __syncthreads();

    const int wave  = tid >> 5;
    const int lane  = tid & 31;
    const int rl    = lane & 15;        // A-row m and B/D-column n
    const int khalf = lane >> 4;        // which K pair this half-wave carries
    const int mbase = khalf * 8;        // D-fragment row base
    const int n0    = wave * 16;        // this wave's N-tile origin

    // ---- gate = sigmoid(pooled @ Wc.T + bc), kept in registers ----
    v8f accg = {};
#pragma unroll 4
    for (int kb = 0; kb < CDIM; kb += 4) {
        const int ka = kb + khalf * 2;
        v2f a, b;
        a.x = pooled[rl * CDIM + ka];
        a.y = pooled[rl * CDIM + ka + 1];
        b.x = Wc[(n0 + rl) * CDIM + ka];
        b.y = Wc[(n0 + rl) * CDIM + ka + 1];
        accg = __builtin_amdgcn_wmma_f32_16x16x4_f32(false, a, false, b,
                                                     (short)0, accg, false, false);
    }
    float gate_frag[8];
    {
        const float bias = bcv[n0 + rl];
#pragma unroll
        for (int v = 0; v < 8; ++v) gate_frag[v] = sigmoidf_(accg[v] + bias);
    }

    // ---- h = relu(pooled @ W1.T + b1), waves 0..7, staged in LDS ----
    if (wave < 8) {
        const int n0h = wave * 16;
        v8f acch = {};
#pragma unroll 4
        for (int kb = 0; kb < CDIM; kb += 4) {
            const int ka = kb + khalf * 2;
            v2f a, b;
            a.x = pooled[rl * CDIM + ka];
            a.y = pooled[rl * CDIM + ka + 1];
            b.x = W1[(n0h + rl) * CDIM + ka];
            b.y = W1[(n0h + rl) * CDIM + ka + 1];
            acch = __builtin_amdgcn_wmma_f32_16x16x4_f32(false, a, false, b,
                                                         (short)0, acch, false, false);
        }
        const float bias = b1[n0h + rl];
#pragma unroll
        for (int v = 0; v < 8; ++v) {
            float hv = fmaxf(acch[v] + bias, 0.0f);
            hbuf[(v + mbase) * HDIM + n0h + rl] = hv;
        }
    }
    __syncthreads();

    // ---- weight = sigmoid(h @ W2.T + b2); s = weight + gate ----
    v8f accw = {};
#pragma unroll 4
    for (int kb = 0; kb < HDIM; kb += 4) {
        const int ka = kb + khalf * 2;
        v2f a, b;
        a.x = hbuf[rl * HDIM + ka];
        a.y = hbuf[rl * HDIM + ka + 1];
        b.x = W2[(n0 + rl) * HDIM + ka];
        b.y = W2[(n0 + rl) * HDIM + ka + 1];
        accw = __builtin_amdgcn_wmma_f32_16x16x4_f32(false, a, false, b,
                                                     (short)0, accw, false, false);
    }
    {
        const float bias = b2[n0 + rl];
#pragma unroll
        for (int v = 0; v < 8; ++v) {
            float wv = sigmoidf_(accw[v] + bias);
            s_out[(v + mbase) * CDIM + n0 + rl] = wv + gate_frag[v];
        }
    }
}

// ---------------------------------------------------------------------------
// Kernel 3: out[b,c,:,:] = x[b,c,:,:] * s[b,c].
// Traverse x in REVERSE (b,c) order: the pool pass streamed x forward through
// the 192 MB L2, so the most-recently-cached tail of x is read first here,
// turning up to ~75% of this pass's reads into L2 hits instead of HBM traffic.
// Output stores are non-temporal (write-once, never re-read) so they don't
// evict x lines we are about to reuse.
// ---------------------------------------------------------------------------
__global__ __launch_bounds__(256) void fsi_scale_kernel(const float* __restrict__ x,
                                                        const float* __restrict__ s,
                                                        float* __restrict__ out) {
    const int bc = (NBC - 1) - blockIdx.x;   // reverse traversal for L2 reuse
    const int t  = threadIdx.x;
    const float sv = s[bc];
    const v4f* __restrict__ x4 = (const v4f*)(x + (size_t)bc * HW_TOTAL);
    v4f* __restrict__ o4       = (v4f*)(out + (size_t)bc * HW_TOTAL);

#pragma unroll
    for (int i = 0; i < 16; ++i) {
        v4f v = x4[i * 256 + t];
        v4f r = v * sv;                      // ext-vector scalar broadcast
        __builtin_nontemporal_store(r, &o4[i * 256 + t]);
    }
}

// ---------------------------------------------------------------------------
extern "C" void kernel_launch(void* const* d_in, const int* in_sizes, int n_in,
                              void* d_out, int out_size, void* d_ws, size_t ws_size,
                              hipStream_t stream) {
    const float* x   = (const float*)d_in[0];
    const float* W1  = (const float*)d_in[1];
    const float* b1  = (const float*)d_in[2];
    const float* W2  = (const float*)d_in[3];
    const float* b2  = (const float*)d_in[4];
    const float* Wc  = (const float*)d_in[5];
    const float* bcv = (const float*)d_in[6];
    float* out = (float*)d_out;

    float* pooled = (float*)d_ws;        // 4096 floats
    float* scale  = pooled + NBC;        // 4096 floats

    fsi_pool_kernel<<<NBC, 256, 0, stream>>>(x, pooled);
    fsi_mlp_kernel<<<1, 512, 0, stream>>>(pooled, W1, b1, W2, b2, Wc, bcv, scale);
    fsi_scale_kernel<<<NBC, 256, 0, stream>>>(x, scale, out);
}